// GoalRGCNLayer_23699629539371
// MI455X (gfx1250) — compile-verified
//
#include <hip/hip_runtime.h>

typedef __attribute__((ext_vector_type(2))) float v2f;
typedef __attribute__((ext_vector_type(8))) float v8f;

#define DIM 64

__device__ __forceinline__ void atomic_add_f32_agent(float* p, float v) {
    // Non-returning, relaxed, device-scope fp32 atomic -> global_atomic_add_f32
    __hip_atomic_fetch_add(p, v, __ATOMIC_RELAXED, __HIP_MEMORY_SCOPE_AGENT);
}
__device__ __forceinline__ int atomic_add_u32_agent(int* p, int v) {
    return __hip_atomic_fetch_add(p, v, __ATOMIC_RELAXED, __HIP_MEMORY_SCOPE_AGENT);
}

// ---------------- zero output + counts ----------------
__global__ void rgcn_zero(float* __restrict__ out, int n, int* __restrict__ counts, int rcnt) {
    int i = blockIdx.x * blockDim.x + threadIdx.x;
    if (i < n) out[i] = 0.0f;
    if (i < rcnt) counts[i] = 0;
}

// ---------------- histogram of relations ----------------
__global__ void rgcn_count(const int* __restrict__ rel, int n, int* __restrict__ counts) {
    int i = blockIdx.x * blockDim.x + threadIdx.x;
    if (i < n) atomic_add_u32_agent(&counts[rel[i]], 1);
}

// ------- exclusive scan over R buckets + build 16-row tile descriptors -------
__global__ void rgcn_build_tiles(const int* __restrict__ counts, int rcnt,
                                 int* __restrict__ cursor, int* __restrict__ tiles,
                                 int* __restrict__ tileCount) {
    if (blockIdx.x == 0 && threadIdx.x == 0) {
        int off = 0, nt = 0;
        for (int r = 0; r < rcnt; ++r) {
            int c = counts[r];
            cursor[r] = off;
            int s = off, rem = c;
            while (rem > 0) {
                tiles[3 * nt + 0] = r;
                tiles[3 * nt + 1] = s;
                tiles[3 * nt + 2] = (rem < 16) ? rem : 16;
                s += 16; rem -= 16; ++nt;
            }
            off += c;
        }
        *tileCount = nt;
    }
}

// ---------------- bucket scatter: sorted[] = node ids grouped by relation ----------------
__global__ void rgcn_scatter_nodes(const int* __restrict__ rel, int n,
                                   int* __restrict__ cursor, int* __restrict__ sorted) {
    int i = blockIdx.x * blockDim.x + threadIdx.x;
    if (i < n) {
        int pos = atomic_add_u32_agent(&cursor[rel[i]], 1);
        sorted[pos] = i;
    }
}

// ---------------- WMMA transform: t[node] = (h[node] @ W[r]) * norm[node] ----------------
// One wave (32 lanes) per tile of up to 16 nodes sharing relation r.
// fp32 path: V_WMMA_F32_16X16X4_F32 chained over K=64 (16 steps) x 4 output col-tiles.
__global__ __launch_bounds__(256) void rgcn_transform_wmma(
    const float* __restrict__ h, const float* __restrict__ norm,
    const float* __restrict__ weight, const int* __restrict__ sorted,
    const int* __restrict__ tiles, const int* __restrict__ tileCount,
    float* __restrict__ t) {

    int wave = (blockIdx.x * blockDim.x + threadIdx.x) >> 5;
    int lane = threadIdx.x & 31;
    int nT = *tileCount;            // uniform across wave
    if (wave >= nT) return;         // wave-uniform branch; EXEC all-1s past here

    int r     = tiles[3 * wave + 0];
    int start = tiles[3 * wave + 1];
    int mcnt  = tiles[3 * wave + 2];

    int half = lane >> 4;           // 0: K={0,1}-of-4, 1: K={2,3}-of-4
    int m    = lane & 15;           // A row (lanes 0-15 and 16-31 both cover M=0..15)

    int rowm = m < mcnt ? m : (mcnt - 1);     // clamp for partial tiles (loads only)
    const float* hrow = h + (size_t)sorted[start + rowm] * DIM;

    // Stage this lane's A fragments for all 16 K-steps: float2 at k = 4*kk + 2*half
    float a[32];
#pragma unroll
    for (int kk = 0; kk < 16; ++kk) {
        v2f av = *(const v2f*)(hrow + 4 * kk + 2 * half);
        a[2 * kk + 0] = av.x;
        a[2 * kk + 1] = av.y;
    }

    const float* W = weight + (size_t)r * DIM * DIM;   // 16 KB, L2-resident, reused by all tiles of r

#pragma unroll
    for (int c = 0; c < 4; ++c) {              // 4 output column tiles of 16
        v8f acc = {};
#pragma unroll
        for (int kk = 0; kk < 16; ++kk) {
            int krow = 4 * kk + 2 * half;
            v2f av; av.x = a[2 * kk]; av.y = a[2 * kk + 1];
            v2f bv;                                         // B 4x16 fragment
            bv.x = W[(size_t)(krow + 0) * DIM + 16 * c + m];
            bv.y = W[(size_t)(krow + 1) * DIM + 16 * c + m];
            acc = __builtin_amdgcn_wmma_f32_16x16x4_f32(
                /*neg_a=*/false, av, /*neg_b=*/false, bv,
                /*c_mod=*/(short)0, acc, /*reuse_a=*/false, /*reuse_b=*/false);
        }
        // D layout: VGPR j -> row (j + 8*half), col = 16*c + (lane&15)
#pragma unroll
        for (int j = 0; j < 8; ++j) {
            int M = j + 8 * half;
            if (M < mcnt) {
                int node = sorted[start + M];
                t[(size_t)node * DIM + 16 * c + m] = acc[j] * norm[node];
            }
        }
    }
}

// ---------------- edge scatter: agg[dst] += t[src] ----------------
// Roofline-critical: E*64 fp32 atomics. Consecutive 64 threads cover one edge's
// row -> coalesced t-row load (L2-resident) + contiguous global_atomic_add_f32.
__global__ void rgcn_edge_scatter(const float* __restrict__ t,
                                  const int* __restrict__ src, const int* __restrict__ dst,
                                  float* __restrict__ out, int e) {
    int i = blockIdx.x * blockDim.x + threadIdx.x;
    if (i < e * DIM) {
        int ed = i >> 6;
        int d  = i & (DIM - 1);
        atomic_add_f32_agent(out + (size_t)dst[ed] * DIM + d,
                             t[(size_t)src[ed] * DIM + d]);
    }
}

// ---------------- bias + relu ----------------
__global__ void rgcn_finalize(float* __restrict__ out, const float* __restrict__ bias, int n) {
    int i = blockIdx.x * blockDim.x + threadIdx.x;
    if (i < n) {
        float v = out[i] + bias[i & (DIM - 1)];
        out[i] = v > 0.0f ? v : 0.0f;
    }
}

extern "C" void kernel_launch(void* const* d_in, const int* in_sizes, int n_in,
                              void* d_out, int out_size, void* d_ws, size_t ws_size,
                              hipStream_t stream) {
    const float* h      = (const float*)d_in[0];
    const float* norm   = (const float*)d_in[1];
    const float* weight = (const float*)d_in[2];
    const float* bias   = (const float*)d_in[3];
    const int*   rel    = (const int*)d_in[4];
    const int*   src    = (const int*)d_in[5];
    const int*   dst    = (const int*)d_in[6];
    float* out = (float*)d_out;

    const int N = in_sizes[1];
    const int E = in_sizes[5];
    const int R = in_sizes[2] / (DIM * DIM);
    const int maxTiles = (N + 15) / 16 + R;

    // workspace layout
    float* t        = (float*)d_ws;                 // N*DIM floats
    int*   counts   = (int*)(t + (size_t)N * DIM);  // R
    int*   cursor   = counts + R;                   // R
    int*   tileCnt  = cursor + R;                   // 1 (+3 pad)
    int*   tiles    = tileCnt + 4;                  // 3*maxTiles
    int*   sorted   = tiles + 3 * maxTiles;         // N

    const int B = 256;
    int nOut = N * DIM;

    rgcn_zero<<<(nOut + B - 1) / B, B, 0, stream>>>(out, nOut, counts, R);
    rgcn_count<<<(N + B - 1) / B, B, 0, stream>>>(rel, N, counts);
    rgcn_build_tiles<<<1, 32, 0, stream>>>(counts, R, cursor, tiles, tileCnt);
    rgcn_scatter_nodes<<<(N + B - 1) / B, B, 0, stream>>>(rel, N, cursor, sorted);

    int waves = maxTiles;                         // one wave per tile
    int blocks = (waves * 32 + B - 1) / B;        // 8 waves per 256-thread block
    rgcn_transform_wmma<<<blocks, B, 0, stream>>>(h, norm, weight, sorted, tiles, tileCnt, t);

    long long escat = (long long)E * DIM;
    rgcn_edge_scatter<<<(int)((escat + B - 1) / B), B, 0, stream>>>(t, src, dst, out, E);
    rgcn_finalize<<<(nOut + B - 1) / B, B, 0, stream>>>(out, bias, nOut);
}